// Custom_trainer_79242146611896
// MI455X (gfx1250) — compile-verified
//
#include <hip/hip_runtime.h>
#include <hip/hip_bf16.h>

// ---------------------------------------------------------------------------
// Sizes from the reference
// ---------------------------------------------------------------------------
constexpr int NN = 16384;  // samples
constexpr int TT = 2048;   // input dim
constexpr int DD = 512;    // latent dim
constexpr int CC = 50;     // classes
constexpr int MS = 16;     // slices for per-class mean reduction

typedef float v2f __attribute__((ext_vector_type(2)));
typedef float v8f __attribute__((ext_vector_type(8)));

// Native fp32 WMMA: D(16x16,f32) = A(16x4,f32) x B(4x16,f32) + C
__device__ __forceinline__ v8f wmma4(v2f a, v2f b, v8f c) {
  return __builtin_amdgcn_wmma_f32_16x16x4_f32(false, a, false, b, (short)0, c,
                                               false, false);
}

// ---------------------------------------------------------------------------
// CDNA5 async global->LDS (ASYNCcnt-tracked), if the toolchain exposes it.
// Builtin signature (from hipcc diagnostic): param0 is
//   '__attribute__((__vector_size__(4*sizeof(int)))) int __device__ *'
// i.e. address_space(1) pointer to a 16-byte int vector; LDS side is the
// address_space(3) analog.
// ---------------------------------------------------------------------------
#if __has_builtin(__builtin_amdgcn_global_load_async_to_lds_b128)
#define GEMM_ASYNC 1
#define GLOBAL_AS __attribute__((address_space(1)))
#define LDS_AS __attribute__((address_space(3)))
typedef int b128_t __attribute__((__vector_size__(4 * sizeof(int))));
__device__ __forceinline__ void async_copy_b128(const void* gsrc, void* ldst) {
  // generic->AS casts via integer round-trip: AS3 pointer == low 32 bits of
  // the generic shared address (hardware truncates flat->LDS the same way).
  __builtin_amdgcn_global_load_async_to_lds_b128(
      (GLOBAL_AS b128_t*)(unsigned long long)gsrc,
      (LDS_AS b128_t*)(unsigned int)(unsigned long long)ldst,
      /*offset=*/0, /*cpol=*/0);
}
#if __has_builtin(__builtin_amdgcn_s_wait_asynccnt)
#define WAIT_ASYNC(n) __builtin_amdgcn_s_wait_asynccnt(n)
#else
#define WAIT_ASYNC(n) asm volatile("s_wait_asynccnt %0" ::"i"(n) : "memory")
#endif
#else
#define GEMM_ASYNC 0
#endif

// ---------------------------------------------------------------------------
// Tiled WMMA GEMM: Cmat[M,Nn] = epilogue(A[M,K] @ B[K,Nn] + bias)
// Block tile 128x64, BK=16. 8 waves, each wave 32x32 (2x2 tiles of 16x16).
// MODE 0: C = tanh(acc + bias)                        (encoder)
// MODE 1: C = acc + bias; partial += |C - aux|        (decoder + rec pinball)
// MODE 2: t = tanh(acc+bias); partial += |t - aux|    (latent pinball, no store)
// ---------------------------------------------------------------------------
template <int MODE>
__global__ __launch_bounds__(256) void gemm_wmma(
    const float* __restrict__ A, const float* __restrict__ B,
    const float* __restrict__ bias, float* __restrict__ Cmat,
    const float* __restrict__ aux, float* __restrict__ partial, int K, int Nn) {
  constexpr int BM = 128, BN = 64, BK = 16;
  constexpr int APAD = 20;  // floats per A row in LDS (80B, 16B aligned)
  constexpr int BPAD = 72;  // floats per B row in LDS (288B, 16B aligned)
#if GEMM_ASYNC
  constexpr int NBUF = 2;  // double-buffered LDS, filled by async DMA
#else
  constexpr int NBUF = 1;  // single buffer + register prefetch
#endif
  __shared__ float ldsA[NBUF][BM * APAD];
  __shared__ float ldsB[NBUF][BK * BPAD];
  __shared__ float red[256];

  const int tid = threadIdx.x;
  const int blockM = blockIdx.y * BM;
  const int blockN = blockIdx.x * BN;

  // ---- staging thread mapping (16B per thread per tile piece) ----
  const int ar0 = tid >> 2;        // 0..63  (A row within half-tile)
  const int ac  = (tid & 3) * 4;   // 0,4,8,12 (A col group)
  const int bkr = tid >> 4;        // 0..15  (B k-row)
  const int bc  = (tid & 15) * 4;  // 0..60  (B col group)

  const float* gA0 = A + (size_t)(blockM + ar0) * K + ac;
  const float* gA1 = gA0 + (size_t)64 * K;

  // ---- wave/lane mapping for WMMA fragments ----
  const int wave = tid >> 5, lane = tid & 31;
  const int wm = wave & 3;   // 0..3 -> M offset wm*32
  const int wn = wave >> 2;  // 0..1 -> N offset wn*32
  const int l16 = lane & 15, hi = lane >> 4;

  // A frag (16x4 f32 layout): lane<16 holds K=k,k+1 ; lane>=16 holds k+2,k+3
  const int aoff0 = (wm * 32 + l16) * APAD + hi * 2;
  const int aoff1 = aoff0 + 16 * APAD;
  // B frag (4x16 f32 layout): lane<16 -> rows k,k+1 ; lane>=16 -> rows k+2,k+3
  const int boff0 = hi * 2 * BPAD + wn * 32 + l16;
  const int boff1 = boff0 + 16;

  v8f acc00 = {0, 0, 0, 0, 0, 0, 0, 0};
  v8f acc01 = {0, 0, 0, 0, 0, 0, 0, 0};
  v8f acc10 = {0, 0, 0, 0, 0, 0, 0, 0};
  v8f acc11 = {0, 0, 0, 0, 0, 0, 0, 0};

  auto compute = [&](const float* bA, const float* bB) {
#pragma unroll
    for (int ks = 0; ks < 4; ++ks) {
      v2f a0 = *(const v2f*)&bA[aoff0 + ks * 4];
      v2f a1 = *(const v2f*)&bA[aoff1 + ks * 4];
      v2f b0, b1;
      b0.x = bB[boff0 + ks * 4 * BPAD];
      b0.y = bB[boff0 + ks * 4 * BPAD + BPAD];
      b1.x = bB[boff1 + ks * 4 * BPAD];
      b1.y = bB[boff1 + ks * 4 * BPAD + BPAD];
      acc00 = wmma4(a0, b0, acc00);
      acc01 = wmma4(a0, b1, acc01);
      acc10 = wmma4(a1, b0, acc10);
      acc11 = wmma4(a1, b1, acc11);
    }
  };

  const int nch = K / BK;

#if GEMM_ASYNC
  // --- async DMA double-buffered pipeline (3 x b128 per thread per chunk) ---
  auto issue = [&](int c, int p) {
    async_copy_b128(gA0 + c * BK, &ldsA[p][ar0 * APAD + ac]);
    async_copy_b128(gA1 + c * BK, &ldsA[p][(ar0 + 64) * APAD + ac]);
    async_copy_b128(B + (size_t)(c * BK + bkr) * Nn + blockN + bc,
                    &ldsB[p][bkr * BPAD + bc]);
  };
  issue(0, 0);
  for (int c = 0; c < nch; ++c) {
    if (c + 1 < nch) {
      issue(c + 1, (c + 1) & 1);  // next chunk stays in flight over compute
      WAIT_ASYNC(3);              // current chunk has landed in LDS
    } else {
      WAIT_ASYNC(0);
    }
    __syncthreads();
    compute(&ldsA[c & 1][0], &ldsB[c & 1][0]);
    __syncthreads();  // everyone done reading before this buffer is refilled
  }
#else
  // --- fallback: register prefetch + single LDS buffer ---
  float4 ra0 = *(const float4*)(gA0);
  float4 ra1 = *(const float4*)(gA1);
  float4 rb  = *(const float4*)(B + (size_t)bkr * Nn + blockN + bc);
  for (int c = 0; c < nch; ++c) {
    *(float4*)&ldsA[0][ar0 * APAD + ac] = ra0;
    *(float4*)&ldsA[0][(ar0 + 64) * APAD + ac] = ra1;
    *(float4*)&ldsB[0][bkr * BPAD + bc] = rb;
    __syncthreads();
    if (c + 1 < nch) {
      ra0 = *(const float4*)(gA0 + (c + 1) * BK);
      ra1 = *(const float4*)(gA1 + (c + 1) * BK);
      rb  = *(const float4*)(B + (size_t)((c + 1) * BK + bkr) * Nn + blockN + bc);
    }
    compute(&ldsA[0][0], &ldsB[0][0]);
    __syncthreads();
  }
#endif

  // ---- epilogue ----
  // C/D layout: VGPR g -> row (g + 8*hi), column = lane%16
  const int col0 = blockN + wn * 32 + l16;
  const int col1 = col0 + 16;
  const int row0 = blockM + wm * 32 + hi * 8;
  const int row1 = row0 + 16;
  const float bv0 = bias[col0];
  const float bv1 = bias[col1];

  float local = 0.0f;
  auto tile = [&](const v8f& acc, int rbase, int col, float bv) {
#pragma unroll
    for (int g = 0; g < 8; ++g) {
      size_t idx = (size_t)(rbase + g) * Nn + col;
      float v = acc[g] + bv;
      if constexpr (MODE == 0) {
        Cmat[idx] = tanhf(v);
      } else if constexpr (MODE == 1) {
        Cmat[idx] = v;
        local += fabsf(v - aux[idx]);
      } else {
        float t = tanhf(v);
        local += fabsf(t - aux[idx]);
      }
    }
  };
  tile(acc00, row0, col0, bv0);
  tile(acc01, row0, col1, bv1);
  tile(acc10, row1, col0, bv0);
  tile(acc11, row1, col1, bv1);

  if constexpr (MODE != 0) {
    red[tid] = local;
    __syncthreads();
    for (int s = 128; s > 0; s >>= 1) {
      if (tid < s) red[tid] += red[tid + s];
      __syncthreads();
    }
    if (tid == 0) partial[blockIdx.y * gridDim.x + blockIdx.x] = red[0];
  }
}

// ---------------------------------------------------------------------------
// Classifier head: per-row softmax -> p_label -> swapped-arg CCE loss.
// cat_labels is one-hot(labels), so loss_i = -(p_l*log(1-e) + (1-p_l)*log(e)).
// ---------------------------------------------------------------------------
__global__ __launch_bounds__(64) void cls_kernel(const float* __restrict__ enc,
                                                 const float* __restrict__ Wc,
                                                 const float* __restrict__ bc,
                                                 const int* __restrict__ labels,
                                                 float* __restrict__ part_cat) {
  __shared__ float row[DD];
  __shared__ float lg[64];
  __shared__ float red[64];
  const int i = blockIdx.x, t = threadIdx.x;
#pragma unroll
  for (int j = 0; j < DD / 64; ++j)
    row[t + j * 64] = enc[(size_t)i * DD + t + j * 64];
  __syncthreads();
  float z = -3.0e38f;
  if (t < CC) {
    z = bc[t];
#pragma unroll 4
    for (int d = 0; d < DD; ++d) z += row[d] * Wc[d * CC + t];
  }
  lg[t] = z;
  red[t] = z;
  __syncthreads();
  for (int s = 32; s > 0; s >>= 1) {
    if (t < s) red[t] = fmaxf(red[t], red[t + s]);
    __syncthreads();
  }
  const float mx = red[0];
  __syncthreads();
  red[t] = (t < CC) ? expf(lg[t] - mx) : 0.0f;
  __syncthreads();
  for (int s = 32; s > 0; s >>= 1) {
    if (t < s) red[t] += red[t + s];
    __syncthreads();
  }
  if (t == 0) {
    float pl = expf(lg[labels[i]] - mx) / red[0];
    const float L1 = logf(1.0f - 1e-7f);  // log(clip hi), f32 const-folded
    const float L0 = logf(1e-7f);         // log(clip lo)
    part_cat[i] = -(pl * L1 + (1.0f - pl) * L0);
  }
}

// ---------------------------------------------------------------------------
// Deterministic per-class means, two phases for parallelism:
//   phase 1: 50 x 16 blocks each scan a 1024-row slice (uniform branch)
//   phase 2: 50 blocks combine the 16 slices
// ---------------------------------------------------------------------------
__global__ __launch_bounds__(512) void means_part_kernel(
    const float* __restrict__ enc, const int* __restrict__ labels,
    float* __restrict__ part_means, float* __restrict__ part_cnt) {
  const int c = blockIdx.x, s = blockIdx.y, d = threadIdx.x;
  const int n0 = s * (NN / MS), n1 = n0 + (NN / MS);
  float acc = 0.0f, cnt = 0.0f;
  for (int n = n0; n < n1; ++n) {
    if (labels[n] == c) {
      acc += enc[(size_t)n * DD + d];
      cnt += 1.0f;
    }
  }
  part_means[((size_t)c * MS + s) * DD + d] = acc;
  if (d == 0) part_cnt[c * MS + s] = cnt;
}

__global__ __launch_bounds__(512) void means_combine_kernel(
    const float* __restrict__ part_means, const float* __restrict__ part_cnt,
    float* __restrict__ means) {
  const int c = blockIdx.x, d = threadIdx.x;
  float acc = 0.0f, cnt = 0.0f;
#pragma unroll
  for (int s = 0; s < MS; ++s) {
    acc += part_means[((size_t)c * MS + s) * DD + d];
    cnt += part_cnt[c * MS + s];
  }
  means[c * DD + d] = acc / fmaxf(cnt, 1.0f);
}

// ---------------------------------------------------------------------------
// Reduce partial arrays to the scalar S added to every output row.
// pinball(eps=0.1,0.9) == 0.9*|diff|; mean-of-means == global mean.
// ---------------------------------------------------------------------------
__device__ __forceinline__ float blk_reduce256(float v, float* red, int t) {
  red[t] = v;
  __syncthreads();
  for (int s = 128; s > 0; s >>= 1) {
    if (t < s) red[t] += red[t + s];
    __syncthreads();
  }
  float r = red[0];
  __syncthreads();
  return r;
}

__global__ __launch_bounds__(256) void scalars_kernel(
    const float* __restrict__ part_rec, int n_rec,
    const float* __restrict__ part_lat, int n_lat,
    const float* __restrict__ part_cat, int n_cat, float* __restrict__ S) {
  __shared__ float red[256];
  const int t = threadIdx.x;
  float a = 0.0f;
  for (int i = t; i < n_rec; i += 256) a += part_rec[i];
  const float s_rec = blk_reduce256(a, red, t);
  a = 0.0f;
  for (int i = t; i < n_lat; i += 256) a += part_lat[i];
  const float s_lat = blk_reduce256(a, red, t);
  a = 0.0f;
  for (int i = t; i < n_cat; i += 256) a += part_cat[i];
  const float s_cat = blk_reduce256(a, red, t);
  if (t == 0) {
    S[0] = 0.9f * s_rec / ((float)NN * (float)TT) +
           0.9f * s_lat / ((float)NN * (float)DD) + s_cat / (float)NN;
  }
}

// ---------------------------------------------------------------------------
// Final per-row output: mean_d (enc - class_mean)^2 + S
// ---------------------------------------------------------------------------
__global__ __launch_bounds__(256) void finalize_kernel(
    const float* __restrict__ enc, const float* __restrict__ means,
    const int* __restrict__ labels, const float* __restrict__ S,
    float* __restrict__ out) {
  __shared__ float red[256];
  const int i = blockIdx.x, t = threadIdx.x;
  const int lbl = labels[i];
  float a = 0.0f;
#pragma unroll
  for (int j = 0; j < DD / 256; ++j) {
    int d = t + j * 256;
    float df = enc[(size_t)i * DD + d] - means[lbl * DD + d];
    a += df * df;
  }
  red[t] = a;
  __syncthreads();
  for (int s = 128; s > 0; s >>= 1) {
    if (t < s) red[t] += red[t + s];
    __syncthreads();
  }
  if (t == 0) out[i] = red[0] * (1.0f / (float)DD) + S[0];
}

// ---------------------------------------------------------------------------
extern "C" void kernel_launch(void* const* d_in, const int* in_sizes, int n_in,
                              void* d_out, int out_size, void* d_ws,
                              size_t ws_size, hipStream_t stream) {
  const float* x      = (const float*)d_in[0];  // [N,T]
  const float* output = (const float*)d_in[1];  // [N,T]
  // d_in[2] = cat_labels (one-hot of labels; handled analytically)
  const int*   labels = (const int*)d_in[3];    // [N]
  const float* W_enc  = (const float*)d_in[4];  // [T,D]
  const float* b_enc  = (const float*)d_in[5];  // [D]
  const float* W_dec  = (const float*)d_in[6];  // [D,T]
  const float* b_dec  = (const float*)d_in[7];  // [T]
  const float* W_cls  = (const float*)d_in[8];  // [D,C]
  const float* b_cls  = (const float*)d_in[9];  // [C]
  float* out = (float*)d_out;

  // workspace layout (floats)
  float* ws = (float*)d_ws;
  float* enc        = ws;                          // N*D
  float* dec        = enc + (size_t)NN * DD;       // N*T
  float* means      = dec + (size_t)NN * TT;       // C*D
  float* part_rec   = means + (size_t)CC * DD;     // 32*128 = 4096
  float* part_lat   = part_rec + 4096;             // 8*128  = 1024
  float* part_cat   = part_lat + 1024;             // N
  float* Sp         = part_cat + NN;               // 8
  float* part_means = Sp + 8;                      // C*MS*D
  float* part_cnt   = part_means + (size_t)CC * MS * DD;  // C*MS

  // 1) encoded = tanh(x @ W_enc + b_enc)
  gemm_wmma<0><<<dim3(DD / 64, NN / 128), 256, 0, stream>>>(
      x, W_enc, b_enc, enc, nullptr, nullptr, TT, DD);
  // 2) classifier loss partials (needs encoded)
  cls_kernel<<<NN, 64, 0, stream>>>(enc, W_cls, b_cls, labels, part_cat);
  // 3) per-class means (needs encoded)
  means_part_kernel<<<dim3(CC, MS), 512, 0, stream>>>(enc, labels, part_means,
                                                      part_cnt);
  means_combine_kernel<<<CC, 512, 0, stream>>>(part_means, part_cnt, means);
  // 4) decoded = encoded @ W_dec + b_dec ; sum |decoded - output| partials
  gemm_wmma<1><<<dim3(TT / 64, NN / 128), 256, 0, stream>>>(
      enc, W_dec, b_dec, dec, output, part_rec, DD, TT);
  // 5) sum |tanh(decoded @ W_enc + b_enc) - encoded| partials (nothing stored)
  gemm_wmma<2><<<dim3(DD / 64, NN / 128), 256, 0, stream>>>(
      dec, W_enc, b_enc, nullptr, enc, part_lat, TT, DD);
  // 6) scalar S = 0.9*mean|rec| + 0.9*mean|lat| + mean(cat_loss)
  scalars_kernel<<<1, 256, 0, stream>>>(part_rec, 4096, part_lat, 1024,
                                        part_cat, NN, Sp);
  // 7) out[i] = mean_d (enc - mean_class)^2 + S
  finalize_kernel<<<NN, 256, 0, stream>>>(enc, means, labels, Sp, out);
}